// Apply_on_single_area_29489245455078
// MI455X (gfx1250) — compile-verified
//
#include <hip/hip_runtime.h>
#include <stdint.h>

// ---------------------------------------------------------------------------
// MI455X (gfx1250) bandwidth-bound fused kernel.
//   B=131072 tiles; per wave: 2 tiles (lanes 0-15 -> tile b0, 16-31 -> b0+1),
//   each lane owns one 16-row column. Inputs staged via the Tensor Data Mover
//   (tensor_load_to_lds) with HW LDS padding for conflict-free transposed
//   reads; outputs repacked in LDS and written as contiguous b128 NT stores.
// ---------------------------------------------------------------------------

#define BN 131072

typedef unsigned int u32x4 __attribute__((ext_vector_type(4)));
typedef int          i32x8 __attribute__((ext_vector_type(8)));
typedef int          i32x4 __attribute__((ext_vector_type(4)));
typedef float        f32x4 __attribute__((ext_vector_type(4)));

#if defined(__HIP_DEVICE_COMPILE__)
#if __has_builtin(__builtin_amdgcn_tensor_load_to_lds) && __has_builtin(__builtin_amdgcn_s_wait_tensorcnt)
#define USE_TDM 1
#endif
#endif

#ifdef USE_TDM
// 1D contiguous TDM load: n_dw dwords global->LDS, inserting 16 pad dwords
// after every (1<<pad_int_enc) interval (enc: 6 -> 128 dw, 7 -> 256 dw).
// D# layout per CDNA5 ISA ch.8 (group0 128b, group1 256b, groups2/3 zero).
// This toolchain exposes the 6-operand builtin:
//   (u32x4 grp0, i32x8 grp1, i32x4 grp2, i32x4 grp3, i32x8 extra, i32 cpol)
__device__ __forceinline__ void tdm_load_1d(unsigned lds_byte, const void* gptr,
                                            unsigned n_dw, int pad_int_enc) {
  unsigned long long ga = (unsigned long long)(uintptr_t)gptr;
  u32x4 g0;
  g0.x = 1u;                                                  // count=1 (valid user D#)
  g0.y = lds_byte;                                            // lds_addr (bytes)
  g0.z = (unsigned)ga;                                        // global_addr[31:0]
  g0.w = (unsigned)((ga >> 32) & 0x01ffffffu) | (2u << 30);   // global_addr[56:32] | type=2
  i32x8 g1;
  g1[0] = (2 << 16)                 // data_size = 4 bytes
        | (1 << 20)                 // pad_enable (LDS dest padding, load only)
        | (pad_int_enc << 22)       // pad_interval
        | (15 << 25);               // pad_amount enc 15 -> 16 dwords
  g1[1] = (int)((n_dw & 0xffffu) << 16);     // tensor_dim0[15:0] @ bits 63:48
  g1[2] = (int)((n_dw >> 16) | (1u << 16));  // tensor_dim0[31:16]; tensor_dim1 = 1
  g1[3] = (int)((n_dw & 0xffffu) << 16);     // tensor_dim1 hi = 0; tile_dim0 = n_dw
  g1[4] = 0;                                 // tile_dim1 = tile_dim2 = 0 (unused)
  g1[5] = (int)n_dw;                         // tensor_dim0_stride (single row -> unused)
  g1[6] = 0;
  g1[7] = 0;
  i32x4 g2 = {0, 0, 0, 0};
  i32x4 g3 = {0, 0, 0, 0};
  i32x8 g4 = {0, 0, 0, 0, 0, 0, 0, 0};       // extra operand group (zero-filled)
  __builtin_amdgcn_tensor_load_to_lds(g0, g1, g2, g3, g4, 0);
}
#endif

// Butterfly sum across a 16-lane half of the wave32 (xor masks never cross bit4).
__device__ __forceinline__ float hsum16(float v) {
  v += __shfl_xor(v, 1, 32);
  v += __shfl_xor(v, 2, 32);
  v += __shfl_xor(v, 4, 32);
  v += __shfl_xor(v, 8, 32);
  return v;
}

// diamond = pad(get_diamond(15),((0,1),(0,1))): 1 iff |m-7| <= n <= 14-|m-7|.
__device__ __forceinline__ float diamondv(int m, int n) {
  int a = m - 7; a = a < 0 ? -a : a;
  return (n >= a && n <= 14 - a) ? 1.0f : 0.0f;
}

// LDS geometry (dwords):
//   per tile: image 256 (+16 pad), mn 128 (+16), mo 128 (+16), mc 256 (+16)
//   per wave (2 tiles): img 544 | mn 288 | mo 288 | mc 544 = 1664 dw = 6656 B
//   8 waves/block -> 53,248 B LDS (fits easily in 320 KB WGP pool).
__global__ __launch_bounds__(256)
void diamond_mask_kernel(const float* __restrict__ image,
                         const float* __restrict__ mnew,
                         const float* __restrict__ mold,
                         float* __restrict__ out) {
  __shared__ __align__(16) float lds[8 * 1664];

  const int tid  = (int)threadIdx.x;
  const int wave = tid >> 5;
  const int lane = tid & 31;
  const int h    = lane >> 4;   // which tile of the wave's pair
  const int c    = lane & 15;   // owned column
  const int b0   = ((int)blockIdx.x * 8 + wave) * 2;

  float* wls = lds + wave * 1664;
  float* IMG = wls;             // 544 dw
  float* MN  = wls + 544;       // 288 dw
  float* MO  = wls + 832;       // 288 dw
  float* MC  = wls + 1120;      // 544 dw

  // ---- stage this wave's inputs into LDS -------------------------------
#ifdef USE_TDM
  {
    unsigned base  = (unsigned)(unsigned long long)(uintptr_t)lds; // LDS byte offset
    unsigned wbyte = base + (unsigned)(wave * 1664 * 4);
    tdm_load_1d(wbyte +    0u, image + (size_t)b0 * 256, 512, 7); // pad 16dw / 256dw
    tdm_load_1d(wbyte + 2176u, mnew  + (size_t)b0 * 128, 256, 6); // pad 16dw / 128dw
    tdm_load_1d(wbyte + 3328u, mold  + (size_t)b0 * 128, 256, 6);
    __builtin_amdgcn_s_wait_tensorcnt(0);
  }
#else
  {
    const size_t gi0 = (size_t)b0 * 256;
    #pragma unroll
    for (int d = lane; d < 512; d += 32) {
      int phys = d + ((d >> 8) << 4);
      IMG[phys] = image[gi0 + d];
    }
    const size_t gm0 = (size_t)b0 * 128;
    #pragma unroll
    for (int d = lane; d < 256; d += 32) {
      int phys = d + ((d >> 7) << 4);
      MN[phys] = mnew[gm0 + d];
      MO[phys] = mold[gm0 + d];
    }
  }
#endif
  // No barrier anywhere: every LDS word this wave reads was written by this
  // wave (TDM done-signal / in-order DS ops give the required ordering).

  const float* img = IMG + h * 272;
  const float* mnp = MN  + h * 144;
  const float* mop = MO  + h * 144;
  float*       mcL = MC  + h * 272;
  float*       imw = IMG + h * 272;  // reused later for out_image

  // ---- gather column c (conflict-free thanks to the 16-dw skew) --------
  float mn8[8], mo8[8], im[16], mc[16];
  #pragma unroll
  for (int i = 0; i < 8; ++i) { mn8[i] = mnp[i * 16 + c]; mo8[i] = mop[i * 16 + c]; }
  #pragma unroll
  for (int m = 0; m < 16; ++m) im[m] = img[m * 16 + c];
  #pragma unroll
  for (int i = 0; i < 8; ++i) { mc[2 * i] = mn8[i]; mc[2 * i + 1] = mo8[i]; }
  #pragma unroll
  for (int m = 0; m < 16; ++m) mcL[m * 16 + c] = mc[m];   // expose mc columns to neighbors

  // ---- per-lane partials ----------------------------------------------
  float cons = 0.f;
  #pragma unroll
  for (int i = 0; i < 8; ++i) {       // grad of zero-padded mo along axis 0
    float lo = (i > 0) ? mo8[i - 1] : 0.f;
    float hi = (i < 7) ? mo8[i + 1] : 0.f;
    float gr = 0.5f * (hi - lo); gr *= gr;
    cons += fmaxf(mn8[i] - 5.f * (gr + mo8[i]), 0.f);
  }

  float rnd = 0.f, cov = 0.f, smc = 0.f, smi = 0.f, edge = 0.f;
  #pragma unroll
  for (int m = 0; m < 16; ++m) {
    float t = 2.f * mc[m] - 1.f; rnd += t * t;
    float d = mc[m] - diamondv(m, c); cov += 0.5f * d * d;
    smc += mc[m]; smi += mc[m] * im[m];
    float gi = (m == 0) ? im[1] - im[0] : (m == 15 ? im[15] - im[14] : 0.5f * (im[m + 1] - im[m - 1]));
    float gm = (m == 0) ? mc[1] - mc[0] : (m == 15 ? mc[15] - mc[14] : 0.5f * (mc[m + 1] - mc[m - 1]));
    float dg = gi - gm; edge += 0.5f * dg * dg * gi;
  }

  // translated masks: axis0 shifts in-register, axis1 shifts via LDS columns
  float s0 = smc - mc[0] - mc[1];
  float s1 = smc - mc[14] - mc[15];
  float q0 = 0.f, q1 = 0.f, s2 = 0.f, q2 = 0.f, s3 = 0.f, q3 = 0.f;
  #pragma unroll
  for (int m = 0; m < 14; ++m) q0 += im[m] * mc[m + 2];
  #pragma unroll
  for (int m = 2; m < 16; ++m) q1 += im[m] * mc[m - 2];
  float mp2[16], mm2[16];
  #pragma unroll
  for (int m = 0; m < 16; ++m) {
    mp2[m] = (c < 14) ? mcL[m * 16 + c + 2] : 0.f;
    mm2[m] = (c >= 2) ? mcL[m * 16 + c - 2] : 0.f;
    s2 += mp2[m]; q2 += im[m] * mp2[m];
    s3 += mm2[m]; q3 += im[m] * mm2[m];
  }

  // ---- first reduction wave (14 values) --------------------------------
  cons = hsum16(cons); rnd = hsum16(rnd); cov = hsum16(cov);
  smc = hsum16(smc); smi = hsum16(smi); edge = hsum16(edge);
  s0 = hsum16(s0); s1 = hsum16(s1); s2 = hsum16(s2); s3 = hsum16(s3);
  q0 = hsum16(q0); q1 = hsum16(q1); q2 = hsum16(q2); q3 = hsum16(q3);
  const float mean0 = q0 / s0, mean1 = q1 / s1, mean2 = q2 / s2, mean3 = q3 / s3;

  // ---- variance pass (needs means) -------------------------------------
  float v0 = 0.f, v1 = 0.f, v2 = 0.f, v3 = 0.f;
  #pragma unroll
  for (int m = 0; m < 16; ++m) {
    float w, t;
    w = (m < 14) ? mc[m + 2] : 0.f; t = (im[m] * w - mean0) * w; v0 += t * t;
    w = (m >= 2) ? mc[m - 2] : 0.f; t = (im[m] * w - mean1) * w; v1 += t * t;
    w = mp2[m];                     t = (im[m] * w - mean2) * w; v2 += t * t;
    w = mm2[m];                     t = (im[m] * w - mean3) * w; v3 += t * t;
  }
  v0 = hsum16(v0) / s0; v1 = hsum16(v1) / s1;
  v2 = hsum16(v2) / s2; v3 = hsum16(v3) / s3;

  const float mmn  = 0.25f * (mean0 + mean1 + mean2 + mean3);
  const float varm = 0.25f * (mean0 * mean0 + mean1 * mean1 + mean2 * mean2 + mean3 * mean3) - mmn * mmn;
  const float mvv  = 0.25f * (v0 + v1 + v2 + v3);
  const float varv = 0.25f * (v0 * v0 + v1 * v1 + v2 * v2 + v3 * v3) - mvv * mvv;
  const float fvar = 0.5f * (varm + varv);
  const float scl  = smi / smc;

  // ---- scalar outputs ---------------------------------------------------
  float* outS = out + (size_t)BN * 512;
  const int b = b0 + h;
  if (c == 0) {
    outS[0 * (size_t)BN + b] = cons * (1.0f / 128.0f);   // consistency
    outS[1 * (size_t)BN + b] = -rnd * (1.0f / 256.0f);   // rounding
    outS[2 * (size_t)BN + b] = fvar;                     // fvar
    outS[3 * (size_t)BN + b] = edge * (1.0f / 256.0f);   // edge
    outS[4 * (size_t)BN + b] = cov * (1.0f / 256.0f);    // avg_cov
  }

  // out_image into LDS (image data is dead now), then repacked wide stores
  #pragma unroll
  for (int m = 0; m < 16; ++m) imw[m * 16 + c] = mc[m] * scl;

  float* outMC = out;
  float* outIM = out + (size_t)BN * 256;
  const size_t obase = (size_t)b0 * 256;
  #pragma unroll
  for (int g = 0; g < 4; ++g) {
    int d    = g * 128 + lane * 4;       // logical dword in [0,512) of wave slice
    int phys = d + ((d >> 8) << 4);      // skip the 16-dw skew pad
    f32x4 vmc = *(const f32x4*)(MC  + phys);
    f32x4 vim = *(const f32x4*)(IMG + phys);
    __builtin_nontemporal_store(vmc, (f32x4*)(outMC + obase + d)); // contiguous 512B/wave
    __builtin_nontemporal_store(vim, (f32x4*)(outIM + obase + d));
  }
}

extern "C" void kernel_launch(void* const* d_in, const int* in_sizes, int n_in,
                              void* d_out, int out_size, void* d_ws, size_t ws_size,
                              hipStream_t stream) {
  (void)in_sizes; (void)n_in; (void)d_ws; (void)ws_size; (void)out_size;
  const float* image = (const float*)d_in[0];
  const float* mnew  = (const float*)d_in[1];
  const float* mold  = (const float*)d_in[2];
  float* out = (float*)d_out;
  // 16 tiles per 256-thread block (8 waves x 2 tiles) -> 8192 blocks
  diamond_mask_kernel<<<dim3(BN / 16), dim3(256), 0, stream>>>(image, mnew, mold, out);
}